// AttentionLayer_3298534883587
// MI455X (gfx1250) — compile-verified
//
#include <hip/hip_runtime.h>
#include <math.h>

// Problem dims (fixed by reference): B=32, T=30, D=512, L=196
#define BB 32
#define TT 30
#define TP 32   // T padded to WMMA tile multiple; pad rows of alpha are zeroed
#define DD 512
#define LL 196

typedef __attribute__((ext_vector_type(2))) float v2f;
typedef __attribute__((ext_vector_type(8))) float v8f;

#if __has_builtin(__builtin_amdgcn_tanhf)
__device__ __forceinline__ float fast_tanh(float x) { return __builtin_amdgcn_tanhf(x); }
#elif __has_builtin(__builtin_amdgcn_tanh_f32)
__device__ __forceinline__ float fast_tanh(float x) { return __builtin_amdgcn_tanh_f32(x); }
#else
__device__ __forceinline__ float fast_tanh(float x) { return tanhf(x); }
#endif

// ---------------------------------------------------------------------------
// Kernel 1: fused attention scores + softmax -> alpha (zero-padded to TP rows).
// One block per (b, t_padded). Pad blocks (t >= 30) just zero their alpha row.
// Real blocks: stage s = x+wordemb and v_w in LDS; 8 waves, one l per wave
// per iteration with coalesced 128B imgsfeats row loads, tanh+FMA (V_TANH_F32
// path), wave32 butterfly reduce; then block-wide softmax over L=196.
// ---------------------------------------------------------------------------
__global__ __launch_bounds__(256)
void attn_score_softmax_kernel(const float* __restrict__ x,
                               const float* __restrict__ wordemb,
                               const float* __restrict__ imgsfeats,
                               const float* __restrict__ v_w,
                               const float* __restrict__ v_b,
                               float* __restrict__ alpha) {
    __shared__ float s_sh[DD];
    __shared__ float vw_sh[DD];
    __shared__ float e_sh[LL + 4];
    __shared__ float red[256];

    const int btp = blockIdx.x;          // 0..B*TP-1
    const int b   = btp >> 5;            // / TP
    const int t   = btp & (TP - 1);
    const int tid = threadIdx.x;

    // Zero-pad rows t = 30, 31 so the WMMA kernel can load unconditionally.
    if (t >= TT) {
        for (int l = tid; l < LL; l += 256) alpha[(size_t)btp * LL + l] = 0.0f;
        return;
    }

    const int bt = b * TT + t;           // index into (B,T,*) tensors
    const int wave = tid >> 5;
    const int lane = tid & 31;

    // Stage s[b,t,:] and v_w[:] in LDS
    for (int d = tid; d < DD; d += 256) {
        s_sh[d]  = x[(size_t)bt * DD + d] + wordemb[(size_t)bt * DD + d];
        vw_sh[d] = v_w[d];
    }
    __syncthreads();

    const float vb0 = v_b[0];
    const float* __restrict__ yb = imgsfeats + (size_t)b * LL * DD;

    // e[l] = sum_d tanh(s[d] + y[l,d]) * vw[d] + vb
    for (int l = wave; l < LL; l += 8) {
        const float* __restrict__ yr = yb + (size_t)l * DD;
        float p = 0.0f;
        #pragma unroll 4
        for (int d = lane; d < DD; d += 32) {
            p = fmaf(fast_tanh(s_sh[d] + yr[d]), vw_sh[d], p);
        }
        #pragma unroll
        for (int off = 16; off >= 1; off >>= 1)
            p += __shfl_xor(p, off, 32);
        if (lane == 0) e_sh[l] = p + vb0;
    }
    __syncthreads();

    // Block-wide softmax over l (one element per thread; tid >= L inert)
    const float neg_inf = -__builtin_huge_valf();
    float v = (tid < LL) ? e_sh[tid] : neg_inf;

    red[tid] = v;
    __syncthreads();
    #pragma unroll
    for (int s = 128; s >= 1; s >>= 1) {
        if (tid < s) red[tid] = fmaxf(red[tid], red[tid + s]);
        __syncthreads();
    }
    const float m = red[0];
    __syncthreads();

    const float ex = (tid < LL) ? __expf(v - m) : 0.0f;
    red[tid] = ex;
    __syncthreads();
    #pragma unroll
    for (int s = 128; s >= 1; s >>= 1) {
        if (tid < s) red[tid] += red[tid + s];
        __syncthreads();
    }
    const float inv = 1.0f / red[0];

    if (tid < LL) alpha[(size_t)btp * LL + tid] = ex * inv;
}

// ---------------------------------------------------------------------------
// Kernel 2: context[b] = alpha[b] (32x196, rows 30-31 zero) @ imgsfeats[b]
// (196x512) via V_WMMA_F32_16X16X4_F32. One wave per 16x16 output tile;
// 2 M-tiles x 32 N-tiles x 32 batches = 2048 waves = 256 blocks of 8 waves.
// All fragment loads unconditional (alpha zero-padded): A is one b64 load,
// B is two b32 loads, pointer-increment addressing. K = 196 = 49 steps of 4.
//
// Fragment layouts per ISA 7.12.2:
//   A 16x4 f32 (2 VGPRs): M = lane&15; VGPR j holds K = k0 + 2*(lane>=16) + j
//   B 4x16 f32 (2 VGPRs): N = lane&15; VGPR j holds K = k0 + 2*(lane>=16) + j
//   C/D 16x16 f32 (8 VGPRs): VGPR r holds M = r + 8*(lane>=16), N = lane&15
// ---------------------------------------------------------------------------
__global__ __launch_bounds__(256)
void context_wmma_kernel(const float* __restrict__ alpha,
                         const float* __restrict__ imgsfeats,
                         float* __restrict__ out) {
    const int wave = threadIdx.x >> 5;
    const int lane = threadIdx.x & 31;

    const int tile  = blockIdx.x * 8 + wave;  // 0..2047
    const int b     = tile >> 6;              // tiles/batch = 64
    const int r     = tile & 63;
    const int mtile = r >> 5;                 // 0..1
    const int ntile = r & 31;                 // 0..31
    const int m_base = mtile * 16;
    const int n_base = ntile * 16;

    const float* __restrict__ A  = alpha     + (size_t)b * TP * LL; // 32 x 196
    const float* __restrict__ Bm = imgsfeats + (size_t)b * LL * DD; // 196 x 512

    const int mrow  = m_base + (lane & 15);   // 0..31, pad rows are zeros
    const int khalf = (lane >> 4) << 1;       // 0 or 2
    const int ncol  = n_base + (lane & 15);

    const float* pA = A  + (size_t)mrow * LL + khalf;   // advance by 4 per step
    const float* pB = Bm + (size_t)khalf * DD + ncol;   // advance by 4*DD

    v8f acc = {};
    #pragma unroll 7
    for (int kk = 0; kk < 49; ++kk) {
        const v2f a = *(const v2f*)pA;        // K = k0+khalf, k0+khalf+1 (8B aligned)
        v2f bf;
        bf.x = pB[0];                          // (K = k0+khalf,   N = ncol)
        bf.y = pB[DD];                         // (K = k0+khalf+1, N = ncol)

        acc = __builtin_amdgcn_wmma_f32_16x16x4_f32(
            /*neg_a=*/false, a, /*neg_b=*/false, bf,
            /*c_mod=*/(short)0, acc, /*reuse_a=*/false, /*reuse_b=*/false);

        pA += 4;
        pB += 4 * DD;
    }

    // Store D fragment (skip padded rows M >= 30)
    #pragma unroll
    for (int rr = 0; rr < 8; ++rr) {
        const int M = m_base + rr + ((lane >> 4) << 3);
        if (M < TT) out[(size_t)b * TT * DD + (size_t)M * DD + ncol] = acc[rr];
    }
}

extern "C" void kernel_launch(void* const* d_in, const int* in_sizes, int n_in,
                              void* d_out, int out_size, void* d_ws, size_t ws_size,
                              hipStream_t stream) {
    const float* x        = (const float*)d_in[0];
    const float* wordemb  = (const float*)d_in[1];
    const float* imgs     = (const float*)d_in[2];
    const float* v_w      = (const float*)d_in[3];
    const float* v_b      = (const float*)d_in[4];
    float*       out      = (float*)d_out;
    float*       alpha    = (float*)d_ws;   // B*TP*LL floats = ~784 KB (zero-padded)

    // Kernel 1: 1024 blocks (one per (b, padded t)), 8 waves each
    attn_score_softmax_kernel<<<dim3(BB * TP), dim3(256), 0, stream>>>(
        x, wordemb, imgs, v_w, v_b, alpha);

    // Kernel 2: 2048 WMMA tiles / 8 waves per block = 256 blocks
    context_wmma_kernel<<<dim3(256), dim3(256), 0, stream>>>(alpha, imgs, out);
}